// GMMConvNet_16561393893736
// MI455X (gfx1250) — compile-verified
//
#include <hip/hip_runtime.h>
#include <hip/hip_bf16.h>

typedef __attribute__((ext_vector_type(2))) float v2f;
typedef __attribute__((ext_vector_type(8))) float v8f;

#define KKER 15      // number of Gaussian kernels
#define PDIM 3       // pseudo-coordinate dim

// GEMM block tile
#define BM 32
#define BN 64
#define BK 32
#define AST (BK + 4)   // As row stride  (As[BM][AST])
#define BST (BK + 4)   // Bs row stride  (Bs[BN][BST], B stored transposed)

// ---------------------------------------------------------------------------
// GEMM: C[M,Ncols] = A[M,Kdim] @ B[Kdim,Ncols], fp32 via V_WMMA_F32_16X16X4_F32
// 256 threads = 8 waves arranged 2(M) x 4(N); each wave owns a 16x16 tile.
// A and B chunks are staged in LDS; interior chunks take an unguarded,
// coalesced fast path, boundary chunks a guarded path. The WMMA loop itself
// is branch-free: ds_load_b64 + v_wmma only (EXEC all-1s as WMMA requires).
// ---------------------------------------------------------------------------
__global__ __launch_bounds__(256) void gemm_f32_wmma(
    const float* __restrict__ A, const float* __restrict__ B,
    float* __restrict__ C, int M, int Kdim, int Ncols) {
  __shared__ float As[BM * AST];
  __shared__ float Bs[BN * BST];

  const int mBlk = blockIdx.x * BM;
  const int nBlk = blockIdx.y * BN;
  const int wave = threadIdx.x >> 5;
  const int lane = threadIdx.x & 31;
  const int half = lane >> 4;     // 0: lanes 0-15, 1: lanes 16-31
  const int l16  = lane & 15;
  const int mw = wave >> 2;       // 0..1  (M sub-tile)
  const int nw = wave & 3;        // 0..3  (N sub-tile)

  const float* __restrict__ Asrow = &As[(mw * 16 + l16) * AST];
  const float* __restrict__ Bsrow = &Bs[(nw * 16 + l16) * BST];

  const bool fullM = (mBlk + BM <= M);
  const bool fullN = (nBlk + BN <= Ncols);

  // Per-thread staging coordinates (fixed across chunks).
  const int ar = threadIdx.x >> 5;           // A row 0..7  (+8 per step of 256)
  const int ak = threadIdx.x & (BK - 1);     // A col 0..31
  const int bk = threadIdx.x >> 6;           // B row 0..3  (+4 per step of 256)
  const int bc = threadIdx.x & (BN - 1);     // B col 0..63

  v8f acc = {};

  for (int k0 = 0; k0 < Kdim; k0 += BK) {
    const bool fullK = (k0 + BK <= Kdim);

    // ---- Stage A chunk [BM x BK] ----
    if (fullM && fullK) {
      const float* __restrict__ Ap = A + (size_t)(mBlk + ar) * Kdim + k0 + ak;
#pragma unroll
      for (int s = 0; s < BM / 8; ++s)       // 4 rows per thread (stride 8)
        As[(ar + 8 * s) * AST + ak] = Ap[(size_t)(8 * s) * Kdim];
    } else {
      for (int idx = threadIdx.x; idx < BM * BK; idx += 256) {
        const int r = idx >> 5;
        const int k = idx & (BK - 1);
        const int m = mBlk + r;
        const int kk = k0 + k;
        As[r * AST + k] = (m < M && kk < Kdim) ? A[(size_t)m * Kdim + kk] : 0.0f;
      }
    }
    // ---- Stage B chunk [BK x BN], transposed into Bs[col][k] ----
    if (fullN && fullK) {
      const float* __restrict__ Bp = B + (size_t)(k0 + bk) * Ncols + nBlk + bc;
#pragma unroll
      for (int s = 0; s < BK / 4; ++s)       // 8 rows per thread (stride 4)
        Bs[bc * BST + bk + 4 * s] = Bp[(size_t)(4 * s) * Ncols];
    } else {
      for (int idx = threadIdx.x; idx < BK * BN; idx += 256) {
        const int k = idx >> 6;
        const int c = idx & (BN - 1);
        const int kk = k0 + k;
        const int col = nBlk + c;
        Bs[c * BST + k] = (kk < Kdim && col < Ncols) ? B[(size_t)kk * Ncols + col] : 0.0f;
      }
    }
    __syncthreads();

    // 8 branch-free WMMA steps per chunk.
    // Fragment layout per ISA: VGPR0 = K{0|2}, VGPR1 = K{1|3} split by lane half.
#pragma unroll
    for (int kk = 0; kk < BK; kk += 4) {
      const int ka = kk + 2 * half;
      v2f a, b;
      a[0] = Asrow[ka];
      a[1] = Asrow[ka + 1];
      b[0] = Bsrow[ka];
      b[1] = Bsrow[ka + 1];
      acc = __builtin_amdgcn_wmma_f32_16x16x4_f32(
          /*neg_a=*/false, a, /*neg_b=*/false, b,
          /*c_mod=*/(short)0, acc, /*reuse_a=*/false, /*reuse_b=*/false);
    }
    __syncthreads();
  }

  // Store D: VGPR j holds row (j + 8*half), column l16 (per ISA C/D layout).
  const int col = nBlk + nw * 16 + l16;
  if (col < Ncols) {
#pragma unroll
    for (int j = 0; j < 8; ++j) {
      const int m = mBlk + mw * 16 + j + 8 * half;
      if (m < M) C[(size_t)m * Ncols + col] = acc[j];
    }
  }
}

// ---------------------------------------------------------------------------
// Per-edge Gaussian mixture weights: w[e,k] = exp(sum_d -0.5*diff^2/(eps+s^2))
// ---------------------------------------------------------------------------
__global__ void edge_weights_kernel(const float* __restrict__ edge_attr,
                                    const float* __restrict__ mu,
                                    const float* __restrict__ sigma,
                                    float* __restrict__ w, int E) {
  const int e = blockIdx.x * blockDim.x + threadIdx.x;
  if (e >= E) return;
  float p[PDIM];
#pragma unroll
  for (int d = 0; d < PDIM; ++d) p[d] = edge_attr[(size_t)e * PDIM + d];
#pragma unroll
  for (int k = 0; k < KKER; ++k) {
    float s = 0.0f;
#pragma unroll
    for (int d = 0; d < PDIM; ++d) {
      const float diff = p[d] - mu[k * PDIM + d];
      const float sg = sigma[k * PDIM + d];
      s += -0.5f * diff * diff / (1e-15f + sg * sg);
    }
    w[(size_t)e * KKER + k] = expf(s);
  }
}

__global__ void zero_kernel(float* __restrict__ p, size_t n) {
  const size_t i = (size_t)blockIdx.x * blockDim.x + threadIdx.x;
  if (i < n) p[i] = 0.0f;
}

__global__ void count_kernel(const int* __restrict__ dst, float* __restrict__ cnt, int E) {
  const int e = blockIdx.x * blockDim.x + threadIdx.x;
  if (e >= E) return;
  atomicAdd(&cnt[dst[e]], 1.0f);
}

// ---------------------------------------------------------------------------
// Edge aggregation: one thread per (edge, channel).
// acc = sum_k w[e,k] * h[src[e], k, c]; atomicAdd into agg[dst[e], c].
// h (<= 195 MB per layer) is L2-resident on MI455X (192 MB L2), so the
// random per-edge gathers hit L2 rather than HBM.
// ---------------------------------------------------------------------------
__global__ void aggregate_kernel(const float* __restrict__ h,
                                 const float* __restrict__ w,
                                 const int* __restrict__ src,
                                 const int* __restrict__ dst,
                                 float* __restrict__ agg, int E, int co) {
  const long long idx = (long long)blockIdx.x * blockDim.x + threadIdx.x;
  const int e = (int)(idx / co);
  if (e >= E) return;
  const int c = (int)(idx - (long long)e * co);
  const int s = src[e];
  const int d = dst[e];
  const float* __restrict__ wp = &w[(size_t)e * KKER];
  const float* __restrict__ hp = &h[(size_t)s * KKER * co + c];
  float acc = 0.0f;
#pragma unroll
  for (int k = 0; k < KKER; ++k) acc += wp[k] * hp[(size_t)k * co];
  atomicAdd(&agg[(size_t)d * co + c], acc);
}

// ---------------------------------------------------------------------------
// out = agg/max(cnt,1) + rootproj + bias, then (layers 0..10) ELU + BatchNorm
// ---------------------------------------------------------------------------
__global__ void finalize_kernel(const float* __restrict__ agg,
                                const float* __restrict__ cnt,
                                const float* __restrict__ rootp,
                                const float* __restrict__ bias,
                                const float* __restrict__ gamma,
                                const float* __restrict__ beta,
                                const float* __restrict__ mean,
                                const float* __restrict__ var,
                                float* __restrict__ out,
                                int N, int co, int apply_act) {
  const long long idx = (long long)blockIdx.x * blockDim.x + threadIdx.x;
  const int n = (int)(idx / co);
  if (n >= N) return;
  const int c = (int)(idx - (long long)n * co);
  float v = agg[idx] / fmaxf(cnt[n], 1.0f) + rootp[idx] + bias[c];
  if (apply_act) {
    v = (v > 0.0f) ? v : (expf(v) - 1.0f);                              // ELU
    v = (v - mean[c]) * (gamma[c] * rsqrtf(var[c] + 1e-5f)) + beta[c];  // BN (eval)
  }
  out[idx] = v;
}

// ---------------------------------------------------------------------------
// Host-side orchestration
// ---------------------------------------------------------------------------
static inline size_t align256(size_t x) { return (x + 255) & ~(size_t)255; }

extern "C" void kernel_launch(void* const* d_in, const int* in_sizes, int n_in,
                              void* d_out, int out_size, void* d_ws, size_t ws_size,
                              hipStream_t stream) {
  static const int DIMS[12][2] = {{200, 75}, {75, 100}, {100, 100}, {100, 100},
                                  {100, 75}, {75, 75}, {75, 50}, {50, 50},
                                  {50, 50}, {50, 50}, {50, 50}, {50, 50}};
  const int N = in_sizes[0] / 200;   // 32492
  const int E = in_sizes[1] / 2;     // 194952

  const float* x_in      = (const float*)d_in[0];
  const int*   src       = (const int*)d_in[1];
  const int*   dst       = src + E;
  const float* edge_attr = (const float*)d_in[2];
  // params: d_in[3 + 5*i + {g,mu,sigma,root,bias}], bn: d_in[63 + 4*i + {gamma,beta,mean,var}]
  const int PBASE = 3, BNBASE = 3 + 5 * 12;

  // Carve workspace
  const int COMAX = 100;
  char* ws = (char*)d_ws;
  size_t off = 0;
  float* d_w    = (float*)(ws + off); off = align256(off + (size_t)E * KKER * sizeof(float));
  float* d_h    = (float*)(ws + off); off = align256(off + (size_t)N * KKER * COMAX * sizeof(float));
  float* d_root = (float*)(ws + off); off = align256(off + (size_t)N * COMAX * sizeof(float));
  float* d_agg  = (float*)(ws + off); off = align256(off + (size_t)N * COMAX * sizeof(float));
  float* d_cnt  = (float*)(ws + off); off = align256(off + (size_t)N * sizeof(float));
  float* d_act0 = (float*)(ws + off); off = align256(off + (size_t)N * COMAX * sizeof(float));
  float* d_act1 = (float*)(ws + off); off = align256(off + (size_t)N * COMAX * sizeof(float));
  (void)ws_size; (void)n_in; (void)out_size;

  // In-degree counts (shared by all layers)
  {
    int blocks = (N + 255) / 256;
    zero_kernel<<<blocks, 256, 0, stream>>>(d_cnt, (size_t)N);
    count_kernel<<<(E + 255) / 256, 256, 0, stream>>>(dst, d_cnt, E);
  }

  const float* act = x_in;
  for (int i = 0; i < 12; ++i) {
    const int ci = DIMS[i][0], co = DIMS[i][1];
    const int Nh = KKER * co;
    const float* g     = (const float*)d_in[PBASE + 5 * i + 0];
    const float* mu    = (const float*)d_in[PBASE + 5 * i + 1];
    const float* sigma = (const float*)d_in[PBASE + 5 * i + 2];
    const float* root  = (const float*)d_in[PBASE + 5 * i + 3];
    const float* bias  = (const float*)d_in[PBASE + 5 * i + 4];

    // 1) per-edge Gaussian weights
    edge_weights_kernel<<<(E + 255) / 256, 256, 0, stream>>>(edge_attr, mu, sigma, d_w, E);

    // 2) h = act @ g  [N, K*co]   (fp32 WMMA)
    {
      dim3 grid((N + BM - 1) / BM, (Nh + BN - 1) / BN);
      gemm_f32_wmma<<<grid, 256, 0, stream>>>(act, g, d_h, N, ci, Nh);
    }
    // 3) rootp = act @ root  [N, co]
    {
      dim3 grid((N + BM - 1) / BM, (co + BN - 1) / BN);
      gemm_f32_wmma<<<grid, 256, 0, stream>>>(act, root, d_root, N, ci, co);
    }

    // 4) zero agg, then gather/weight/scatter-add over edges
    {
      size_t na = (size_t)N * co;
      zero_kernel<<<(int)((na + 255) / 256), 256, 0, stream>>>(d_agg, na);
      long long work = (long long)E * co;
      aggregate_kernel<<<(int)((work + 255) / 256), 256, 0, stream>>>(
          d_h, d_w, src, dst, d_agg, E, co);
    }

    // 5) mean-normalize + root + bias (+ ELU + BN for layers 0..10)
    {
      float* outp = (i == 11) ? (float*)d_out : ((i & 1) ? d_act1 : d_act0);
      const float *gamma = nullptr, *beta = nullptr, *mean = nullptr, *var = nullptr;
      int apply_act = (i < 11) ? 1 : 0;
      if (apply_act) {
        gamma = (const float*)d_in[BNBASE + 4 * i + 0];
        beta  = (const float*)d_in[BNBASE + 4 * i + 1];
        mean  = (const float*)d_in[BNBASE + 4 * i + 2];
        var   = (const float*)d_in[BNBASE + 4 * i + 3];
      }
      long long work = (long long)N * co;
      finalize_kernel<<<(int)((work + 255) / 256), 256, 0, stream>>>(
          d_agg, d_cnt, d_root, bias, gamma, beta, mean, var, outp, N, co, apply_act);
      act = outp;
    }
  }
}